// SwinLSTM_12867722019354
// MI455X (gfx1250) — compile-verified
//
#include <hip/hip_runtime.h>
#include <hip/hip_bf16.h>

typedef __attribute__((ext_vector_type(2))) float v2f;
typedef __attribute__((ext_vector_type(8))) float v8f;

#define HC   128
#define XC   64
#define CC   192
#define HW   16384      // 128*128
#define BATCH 16
#define NPLANE ((size_t)HC * HW)          // 2,097,152
#define HALF_OUT ((size_t)BATCH * NPLANE) // 33,554,432

__device__ __forceinline__ float sigf(float x) {
    return 1.0f / (1.0f + __expf(-x));
}

// ---------------------------------------------------------------- zero stats
__global__ void swin_zero_kernel(float* p, int n) {
    int i = blockIdx.x * blockDim.x + threadIdx.x;
    if (i < n) p[i] = 0.0f;
}

// ---------------------------------------------- depthwise 7x7 conv + GN1 stats
// grid: (4, 4, B*CC), block: (32, 8)
__global__ void swin_dwconv_kernel(const float* __restrict__ x,
                                   const float* __restrict__ h,
                                   const float* __restrict__ dww,
                                   const float* __restrict__ dwb,
                                   float* __restrict__ z,
                                   float* __restrict__ stats1) {
    __shared__ float tile[38 * 38];
    __shared__ float wf[49];
    __shared__ float red[256];

    const int bc = blockIdx.z;
    const int b = bc / CC;
    const int c = bc % CC;
    const int tid = threadIdx.y * 32 + threadIdx.x;

    const float* src = (c < XC) ? (x + ((size_t)b * XC + c) * HW)
                                : (h + ((size_t)b * HC + (c - XC)) * HW);
    if (tid < 49) wf[tid] = dww[c * 49 + tid];

    const int ox0 = blockIdx.x * 32, oy0 = blockIdx.y * 32;
    for (int t = tid; t < 38 * 38; t += 256) {
        int ly = t / 38, lx = t % 38;
        int gy = oy0 + ly - 3, gx = ox0 + lx - 3;
        float v = 0.0f;
        if (gy >= 0 && gy < 128 && gx >= 0 && gx < 128) v = src[gy * 128 + gx];
        tile[t] = v;
    }
    __syncthreads();

    const float bias = dwb[c];
    float lsum = 0.0f, lsq = 0.0f;
    float* zpl = z + (size_t)bc * HW;
#pragma unroll
    for (int i = 0; i < 4; ++i) {
        int ty = threadIdx.y + 8 * i;
        int tx = threadIdx.x;
        float acc = bias;
#pragma unroll
        for (int ky = 0; ky < 7; ++ky)
#pragma unroll
            for (int kx = 0; kx < 7; ++kx)
                acc += wf[ky * 7 + kx] * tile[(ty + ky) * 38 + (tx + kx)];
        zpl[(oy0 + ty) * 128 + (ox0 + tx)] = acc;
        lsum += acc;
        lsq += acc * acc;
    }

    red[tid] = lsum; __syncthreads();
    for (int s = 128; s > 0; s >>= 1) { if (tid < s) red[tid] += red[tid + s]; __syncthreads(); }
    float bsum = red[0]; __syncthreads();
    red[tid] = lsq; __syncthreads();
    for (int s = 128; s > 0; s >>= 1) { if (tid < s) red[tid] += red[tid + s]; __syncthreads(); }
    if (tid == 0) {
        int g = c / 48;
        atomicAdd(&stats1[(b * 4 + g) * 2 + 0], bsum);
        atomicAdd(&stats1[(b * 4 + g) * 2 + 1], red[0]);
    }
}

// ------------------------------------------------------------- finalize stats
__global__ void swin_finalize_kernel(const float* __restrict__ stats,
                                     float* __restrict__ mr, int count, float inv_n) {
    int t = threadIdx.x;
    if (t < count) {
        float s = stats[t * 2], ss = stats[t * 2 + 1];
        float mu = s * inv_n;
        float var = ss * inv_n - mu * mu;
        mr[t * 2 + 0] = mu;
        mr[t * 2 + 1] = rsqrtf(var + 1e-5f);
    }
}

// --------------------------- fused WMMA pointwise GEMM + LSTM gates + GN2 stats
// grid: (128, 8, B), block: 256 (8 waves). Wave = 16 hc-channels x 16 columns,
// computing i/f/o/g tiles together via V_WMMA_F32_16X16X4_F32, K = 192.
#define WSTR 196   // padded LDS row stride (floats): conflict-free, 8B aligned
__global__ void swin_gemm_lstm_kernel(const float* __restrict__ zws,
                                      const float* __restrict__ pww,
                                      const float* __restrict__ pwb,
                                      const float* __restrict__ cin,
                                      const float* __restrict__ mr1,
                                      float* __restrict__ cnew_out,
                                      float* __restrict__ sigo,
                                      float* __restrict__ stats2) {
    __shared__ float ldsW[64 * WSTR];
    __shared__ float red[256];

    const int tid = threadIdx.x;
    const int b = blockIdx.z;
    const int c0 = blockIdx.y * 16;

    // Stage the 4-gate pw_w strip (64 rows x 192 K) into LDS.
    for (int t = tid; t < 64 * 192; t += 256) {
        int row = t / 192, k = t % 192;
        int gi = row >> 4, m = row & 15;
        ldsW[row * WSTR + k] = pww[((size_t)(gi * HC + c0 + m)) * CC + k];
    }

    // GroupNorm-1 constants, folded: z_norm = z*ga + gb
    float ga[4], gb[4];
#pragma unroll
    for (int g = 0; g < 4; ++g) {
        float mu = mr1[(b * 4 + g) * 2], rs = mr1[(b * 4 + g) * 2 + 1];
        ga[g] = rs;
        gb[g] = -mu * rs;
    }
    __syncthreads();

    const int wave = tid >> 5, lane = tid & 31;
    const int m = lane & 15, hi = lane >> 4;
    const int n0 = (blockIdx.x * 8 + wave) * 16;
    const int n = n0 + m;
    const float* zb = zws + (size_t)b * CC * HW;

    v8f acc[4] = {v8f{}, v8f{}, v8f{}, v8f{}};

    for (int kk = 0; kk < 48; ++kk) {
        const int kbase = kk * 4;
        const int g = kk / 12;          // 12 k-steps per 48-channel group
        const int kc = kbase + 2 * hi;  // this lane's K pair base
        const float* zp = zb + (size_t)kc * HW + n;
        if (kk < 44)                    // stream-prefetch z 4 k-steps ahead
            __builtin_prefetch(zp + (size_t)16 * HW, 0, 0);

        v2f bf;
        bf.x = zp[0] * ga[g] + gb[g];
        bf.y = zp[HW] * ga[g] + gb[g];
#pragma unroll
        for (int gi = 0; gi < 4; ++gi) {
            v2f af = *(const v2f*)&ldsW[(gi * 16 + m) * WSTR + kc];
            acc[gi] = __builtin_amdgcn_wmma_f32_16x16x4_f32(
                false, af, false, bf, (short)0, acc[gi], false, false);
        }
    }

    // Epilogue: bias, gate math, c_new, sigma(o), GN2 partial stats.
    float lsum = 0.0f, lsq = 0.0f;
#pragma unroll
    for (int r = 0; r < 8; ++r) {
        int ch = c0 + r + 8 * hi;
        float vi = acc[0][r] + pwb[0 * HC + ch];
        float vf = acc[1][r] + pwb[1 * HC + ch];
        float vo = acc[2][r] + pwb[2 * HC + ch];
        float vg = acc[3][r] + pwb[3 * HC + ch];
        size_t oidx = ((size_t)b * HC + ch) * HW + n;
        float cold = cin[oidx];
        float cn = sigf(vf) * cold + sigf(vi) * tanhf(vg);
        cnew_out[oidx] = cn;
        sigo[oidx] = sigf(vo);
        lsum += cn;
        lsq += cn * cn;
    }

    red[tid] = lsum; __syncthreads();
    for (int s = 128; s > 0; s >>= 1) { if (tid < s) red[tid] += red[tid + s]; __syncthreads(); }
    float bs = red[0]; __syncthreads();
    red[tid] = lsq; __syncthreads();
    for (int s = 128; s > 0; s >>= 1) { if (tid < s) red[tid] += red[tid + s]; __syncthreads(); }
    if (tid == 0) {
        atomicAdd(&stats2[b * 2 + 0], bs);
        atomicAdd(&stats2[b * 2 + 1], red[0]);
    }
}

// ------------------------------------------------------------------ h_new pass
__global__ void swin_hnew_kernel(const float* __restrict__ cnew,
                                 const float* __restrict__ sigo,
                                 const float* __restrict__ mr2,
                                 const float* __restrict__ gamma,
                                 const float* __restrict__ beta,
                                 float* __restrict__ hout) {
    size_t idx = (size_t)blockIdx.x * blockDim.x + threadIdx.x;
    if (idx >= HALF_OUT) return;
    int b = (int)(idx >> 21);          // 128*16384 = 2^21 per batch
    int ch = (int)((idx >> 14) & 127); // 16384 = 2^14 per plane
    float mu = mr2[b * 2], rs = mr2[b * 2 + 1];
    float cn = cnew[idx];
    float so = sigo[idx];
    hout[idx] = so * tanhf((cn - mu) * rs * gamma[ch] + beta[ch]);
}

// ------------------------------------------------------------------- launcher
extern "C" void kernel_launch(void* const* d_in, const int* in_sizes, int n_in,
                              void* d_out, int out_size, void* d_ws, size_t ws_size,
                              hipStream_t stream) {
    const float* x     = (const float*)d_in[0];
    const float* h     = (const float*)d_in[1];
    const float* c     = (const float*)d_in[2];
    const float* dww   = (const float*)d_in[3];
    const float* dwb   = (const float*)d_in[4];
    const float* pww   = (const float*)d_in[5];
    const float* pwb   = (const float*)d_in[6];
    const float* gamma = (const float*)d_in[7];
    const float* beta  = (const float*)d_in[8];

    float* out  = (float*)d_out;
    float* hnew = out;                 // first tuple element
    float* cnew = out + HALF_OUT;      // second tuple element

    float* ws     = (float*)d_ws;
    float* z      = ws;                               // 192*16*16384 floats
    float* sigo   = ws + (size_t)CC * BATCH * HW;     // 128*16*16384 floats
    float* stats1 = sigo + HALF_OUT;                  // 128
    float* mr1    = stats1 + 128;                     // 128
    float* stats2 = mr1 + 128;                        // 32
    float* mr2    = stats2 + 32;                      // 32

    // 1. zero stat scratch (d_ws is poisoned; must be deterministic per call)
    swin_zero_kernel<<<1, 320, 0, stream>>>(stats1, 320);

    // 2. depthwise conv + GN1 partial stats
    dim3 dgrid(4, 4, BATCH * CC);
    swin_dwconv_kernel<<<dgrid, dim3(32, 8), 0, stream>>>(x, h, dww, dwb, z, stats1);

    // 3. finalize GN1: 16 batches x 4 groups, N = 48*16384
    swin_finalize_kernel<<<1, 64, 0, stream>>>(stats1, mr1, 64, 1.0f / 786432.0f);

    // 4. fused WMMA GEMM + LSTM gates + GN2 partial stats
    dim3 ggrid(128, 8, BATCH);
    swin_gemm_lstm_kernel<<<ggrid, 256, 0, stream>>>(z, pww, pwb, c, mr1,
                                                     cnew, sigo, stats2);

    // 5. finalize GN2: 16 batches, N = 128*16384
    swin_finalize_kernel<<<1, 16, 0, stream>>>(stats2, mr2, 16, 1.0f / 2097152.0f);

    // 6. h_new
    swin_hnew_kernel<<<(unsigned)((HALF_OUT + 255) / 256), 256, 0, stream>>>(
        cnew, sigo, mr2, gamma, beta, hnew);
}